// CustomConv_84301618085957
// MI455X (gfx1250) — compile-verified
//
#include <hip/hip_runtime.h>

typedef __attribute__((ext_vector_type(2))) float v2f;
typedef __attribute__((ext_vector_type(8))) float v8f;

#define N_NODES 50000
#define E_EDGES 800000
#define HDIM 256
#define PDIM 128

// ---------------------------------------------------------------------------
// z0[i,:] = node_embed[node_x[i], :]   (float4 vectorized)
// ---------------------------------------------------------------------------
__global__ void k_gather_embed(const int* __restrict__ node_x,
                               const float* __restrict__ emb,
                               float* __restrict__ z) {
  int gid = blockIdx.x * blockDim.x + threadIdx.x;
  const int total = N_NODES * (HDIM / 4);
  if (gid >= total) return;
  int i = gid / (HDIM / 4);
  int q = gid % (HDIM / 4);
  const float4* src = (const float4*)(emb + (size_t)node_x[i] * HDIM);
  ((float4*)(z + (size_t)i * HDIM))[q] = src[q];
}

__global__ void k_zero_int(int* __restrict__ p, int n) {
  int gid = blockIdx.x * blockDim.x + threadIdx.x;
  if (gid < n) p[gid] = 0;
}

__global__ void k_count_deg(const int* __restrict__ col, int* __restrict__ cnt) {
  int gid = blockIdx.x * blockDim.x + threadIdx.x;
  if (gid < E_EDGES) atomicAdd(&cnt[col[gid]], 1);
}

// deg includes +1 for the self loop; deg >= 1 so no max() needed.
__global__ void k_deg_inv(float* __restrict__ buf) {
  int gid = blockIdx.x * blockDim.x + threadIdx.x;
  if (gid >= N_NODES) return;
  int c = ((const int*)buf)[gid];
  buf[gid] = 1.0f / (float)(c + 1);
}

// agg[i,:] = z[i,:] * deg_inv[i]   (self-loop contribution; also zero-inits agg)
__global__ void k_self_init(const float* __restrict__ z,
                            const float* __restrict__ dinv,
                            float* __restrict__ agg) {
  int gid = blockIdx.x * blockDim.x + threadIdx.x;
  const int total = N_NODES * (HDIM / 4);
  if (gid >= total) return;
  int i = gid / (HDIM / 4);
  int q = gid % (HDIM / 4);
  float s = dinv[i];
  float4 v = ((const float4*)(z + (size_t)i * HDIM))[q];
  v.x *= s; v.y *= s; v.z *= s; v.w *= s;
  ((float4*)(agg + (size_t)i * HDIM))[q] = v;
}

// 64 threads per edge, float4 each: agg[col[e],:] += z[row[e],:] * dinv[col[e]]
__global__ void k_scatter(const int* __restrict__ row,
                          const int* __restrict__ col,
                          const float* __restrict__ dinv,
                          const float* __restrict__ z,
                          float* __restrict__ agg) {
  int gid = blockIdx.x * blockDim.x + threadIdx.x;
  int e = gid >> 6;                 // H/4 = 64 quads per edge
  if (e >= E_EDGES) return;
  int q = gid & 63;
  int r = row[e];
  int c = col[e];
  float s = dinv[c];
  float4 v = ((const float4*)(z + (size_t)r * HDIM))[q];
  float* dst = agg + (size_t)c * HDIM + q * 4;
  atomicAdd(dst + 0, v.x * s);
  atomicAdd(dst + 1, v.y * s);
  atomicAdd(dst + 2, v.z * s);
  atomicAdd(dst + 3, v.w * s);
}

// ---------------------------------------------------------------------------
// out = relu(agg @ Wo + z @ Wr + bo)   [N,256] via f32 WMMA 16x16x4.
// Block = 8 waves; each wave owns one 16x16 output tile; grid.y = H/16.
// Wo/Wr 16-wide column panels staged in LDS (32 KB).
// ---------------------------------------------------------------------------
__global__ void __launch_bounds__(256) k_layer_gemm(
    const float* __restrict__ A,    // agg [N,H]
    const float* __restrict__ Z,    // z   [N,H]
    const float* __restrict__ Wo,   // [H,H]
    const float* __restrict__ Wr,   // [H,H]
    const float* __restrict__ bo,   // [H]
    float* __restrict__ out) {      // [N,H]
  __shared__ float sWo[HDIM * 16];
  __shared__ float sWr[HDIM * 16];
  const int n0 = blockIdx.y * 16;
  for (int idx = threadIdx.x; idx < HDIM * 16; idx += 256) {
    int k = idx >> 4;
    int n = idx & 15;
    sWo[idx] = Wo[k * HDIM + n0 + n];
    sWr[idx] = Wr[k * HDIM + n0 + n];
  }
  __syncthreads();

  const int wave = threadIdx.x >> 5;
  const int lane = threadIdx.x & 31;
  const int tile = blockIdx.x * 8 + wave;        // N/16 = 3125 tiles exactly
  if (tile * 16 >= N_NODES) return;
  const int m0 = tile * 16;

  const int m  = lane & 15;        // A row within tile
  const int kb = (lane >> 4) * 2;  // A/B K sub-offset per ISA f32 layout
  const int n  = lane & 15;        // B/C column within tile

  const float* arow = A + (size_t)(m0 + m) * HDIM + kb;
  const float* zrow = Z + (size_t)(m0 + m) * HDIM + kb;

  v8f c = {};
  #pragma unroll 4
  for (int k = 0; k < HDIM; k += 4) {
    v2f a0 = *(const v2f*)(arow + k);
    v2f b0;
    b0.x = sWo[(k + kb) * 16 + n];
    b0.y = sWo[(k + kb + 1) * 16 + n];
    c = __builtin_amdgcn_wmma_f32_16x16x4_f32(false, a0, false, b0,
                                              (short)0, c, false, false);
    v2f a1 = *(const v2f*)(zrow + k);
    v2f b1;
    b1.x = sWr[(k + kb) * 16 + n];
    b1.y = sWr[(k + kb + 1) * 16 + n];
    c = __builtin_amdgcn_wmma_f32_16x16x4_f32(false, a1, false, b1,
                                              (short)0, c, false, false);
  }

  const float bias = bo[n0 + n];
  const int rbase = m0 + ((lane >> 4) << 3);   // lanes16-31 hold M = r+8
  #pragma unroll
  for (int r = 0; r < 8; ++r) {
    float v = c[r] + bias;
    v = v > 0.0f ? v : 0.0f;
    out[(size_t)(rbase + r) * HDIM + n0 + n] = v;
  }
}

// ---------------------------------------------------------------------------
// Generic single GEMM: out = act(A[M,K] @ W[K,NC] + bias), NC = out stride.
// ACT: 0 = none, 2 = PReLU(alpha from pointer).
// ---------------------------------------------------------------------------
template <int K, int ACT>
__global__ void __launch_bounds__(256) k_gemm(
    const float* __restrict__ A,
    const float* __restrict__ W,      // [K,NC]
    const float* __restrict__ bias,   // [NC]
    const float* __restrict__ alpha_p,
    float* __restrict__ out,          // [N,NC]
    int NC) {
  __shared__ float sW[K * 16];
  const int n0 = blockIdx.y * 16;
  for (int idx = threadIdx.x; idx < K * 16; idx += 256) {
    int k = idx >> 4;
    int n = idx & 15;
    sW[idx] = W[k * NC + n0 + n];
  }
  __syncthreads();

  const int wave = threadIdx.x >> 5;
  const int lane = threadIdx.x & 31;
  const int tile = blockIdx.x * 8 + wave;
  if (tile * 16 >= N_NODES) return;
  const int m0 = tile * 16;

  const int m  = lane & 15;
  const int kb = (lane >> 4) * 2;
  const int n  = lane & 15;

  const float* arow = A + (size_t)(m0 + m) * K + kb;

  v8f c = {};
  #pragma unroll 4
  for (int k = 0; k < K; k += 4) {
    v2f a = *(const v2f*)(arow + k);
    v2f b;
    b.x = sW[(k + kb) * 16 + n];
    b.y = sW[(k + kb + 1) * 16 + n];
    c = __builtin_amdgcn_wmma_f32_16x16x4_f32(false, a, false, b,
                                              (short)0, c, false, false);
  }

  const float bv = bias[n0 + n];
  float alpha = 0.0f;
  if (ACT == 2) alpha = alpha_p[0];
  const int rbase = m0 + ((lane >> 4) << 3);
  #pragma unroll
  for (int r = 0; r < 8; ++r) {
    float v = c[r] + bv;
    if (ACT == 2) v = v > 0.0f ? v : alpha * v;
    out[(size_t)(rbase + r) * NC + n0 + n] = v;
  }
}

// ---------------------------------------------------------------------------
extern "C" void kernel_launch(void* const* d_in, const int* in_sizes, int n_in,
                              void* d_out, int out_size, void* d_ws, size_t ws_size,
                              hipStream_t stream) {
  const int*   node_x     = (const int*)d_in[0];
  // d_in[1] = edge_type (unused by reference math)
  const int*   edge_index = (const int*)d_in[2];
  const int*   row        = edge_index;            // edge_index[0]
  const int*   col        = edge_index + E_EDGES;  // edge_index[1]
  const float* node_embed = (const float*)d_in[3];
  const float* W_out      = (const float*)d_in[4]; // [3,H,H]
  const float* b_out      = (const float*)d_in[5]; // [3,H]
  const float* W_root     = (const float*)d_in[6]; // [3,H,H]
  const float* W1         = (const float*)d_in[7]; // [H,P]
  const float* b1         = (const float*)d_in[8]; // [P]
  const float* prelu_a    = (const float*)d_in[9]; // [1]
  const float* W2         = (const float*)d_in[10];// [P,P]
  const float* b2         = (const float*)d_in[11];// [P]

  float* z_final = (float*)d_out;                       // [N,H]
  float* proj    = z_final + (size_t)N_NODES * HDIM;    // [N,P]

  float* ws   = (float*)d_ws;
  float* dinv = ws;                                     // N (int counts, then 1/deg)
  float* zA   = dinv + N_NODES;                         // [N,H]
  float* zB   = zA + (size_t)N_NODES * HDIM;            // [N,H]
  float* agg  = zB + (size_t)N_NODES * HDIM;            // [N,H]
  float* hbuf = agg + (size_t)N_NODES * HDIM;           // [N,P]

  const int nh4 = N_NODES * (HDIM / 4);

  // 1) Embedding gather
  k_gather_embed<<<(nh4 + 255) / 256, 256, 0, stream>>>(node_x, node_embed, zA);

  // 2) Degrees (edge targets + self loop) -> 1/deg
  k_zero_int<<<(N_NODES + 255) / 256, 256, 0, stream>>>((int*)dinv, N_NODES);
  k_count_deg<<<(E_EDGES + 255) / 256, 256, 0, stream>>>(col, (int*)dinv);
  k_deg_inv<<<(N_NODES + 255) / 256, 256, 0, stream>>>(dinv);

  // 3) Three ClusterGCN layers
  const float* zin[3]  = {zA, zB, zA};
  float*       zout[3] = {zB, zA, z_final};
  dim3 ggrid((N_NODES / 16 + 7) / 8, HDIM / 16);
  const int sc_blocks = (E_EDGES * 64 + 255) / 256;
  for (int l = 0; l < 3; ++l) {
    k_self_init<<<(nh4 + 255) / 256, 256, 0, stream>>>(zin[l], dinv, agg);
    k_scatter<<<sc_blocks, 256, 0, stream>>>(row, col, dinv, zin[l], agg);
    k_layer_gemm<<<ggrid, 256, 0, stream>>>(
        agg, zin[l],
        W_out + (size_t)l * HDIM * HDIM,
        W_root + (size_t)l * HDIM * HDIM,
        b_out + (size_t)l * HDIM,
        zout[l]);
  }

  // 4) h = PReLU(z @ W1 + b1), proj = h @ W2 + b2
  dim3 pgrid((N_NODES / 16 + 7) / 8, PDIM / 16);
  k_gemm<HDIM, 2><<<pgrid, 256, 0, stream>>>(z_final, W1, b1, prelu_a, hbuf, PDIM);
  k_gemm<PDIM, 0><<<pgrid, 256, 0, stream>>>(hbuf, W2, b2, prelu_a, proj, PDIM);
}